// DynamicPointNet_42932493091360
// MI455X (gfx1250) — compile-verified
//
#include <hip/hip_runtime.h>
#include <hip/hip_bf16.h>

// ---------------- problem constants (match reference) ----------------
#define NPTS   2000000
#define NUM_IN 9
#define F1     32
#define F2     64
#define NSEG   100000
#define NTILE  (NPTS / 16)      // 125000 tiles of 16 points
#define BN_EPS 1e-5f

// workspace layout (floats)
#define WS_SUM1 0     // 32
#define WS_SQ1  32    // 32
#define WS_SUM2 64    // 64
#define WS_SQ2  128   // 64
#define WS_SC1  192   // 32
#define WS_SH1  224   // 32
#define WS_SC2  256   // 64
#define WS_SH2  320   // 64
#define WS_ACC_FLOATS 192

typedef float v2f __attribute__((ext_vector_type(2)));
typedef float v8f __attribute__((ext_vector_type(8)));

__device__ __forceinline__ v8f splat8(float b) {
    v8f v;
#pragma unroll
    for (int i = 0; i < 8; ++i) v[i] = b;
    return v;
}

// ---------------------------------------------------------------------
// 0) zero output + stat accumulators
// ---------------------------------------------------------------------
__global__ void zero_kernel(float* __restrict__ out, int n_out, float* __restrict__ ws) {
    int i = blockIdx.x * 256 + threadIdx.x;
    if (i < n_out) out[i] = 0.0f;
    if (i < WS_ACC_FLOATS) ws[i] = 0.0f;
}

// ---------------------------------------------------------------------
// 1) layer-1 pre-activation stats via WMMA.
//    A = points tile (16x12, K rows 9..11 zero), B = w1 (12x32, rows
//    9..11 zero), C = b1 splat. D layout puts feature on the lane, so
//    the per-feature sum/sumsq reduction is a register accumulate.
// ---------------------------------------------------------------------
__global__ void __launch_bounds__(256, 2)
stats1_kernel(const float* __restrict__ points,
              const float* __restrict__ w1,
              const float* __restrict__ b1,
              float* __restrict__ ws) {
    __shared__ float sstat[2 * F1];

    int tid  = threadIdx.x;
    int lane = tid & 31;
    int ml   = lane & 15;
    bool hi  = lane >= 16;
    int hi2  = hi ? 2 : 0;
    int ko   = hi2;

    if (tid < 2 * F1) sstat[tid] = 0.0f;
    __syncthreads();

    // loop-invariant B fragments for w1 (12x32, zero-padded K)
    v2f Bw1[2][3];
    float bias1[2];
#pragma unroll
    for (int t = 0; t < 2; ++t) {
        int n1 = t * 16 + ml;
        bias1[t] = b1[n1];
#pragma unroll
        for (int c = 0; c < 3; ++c) {
            int k0 = 4 * c + hi2;                 // 0,4,8 (lo) / 2,6,10 (hi)
            int ka = k0 < 8 ? k0 : 8;
            int kb = (k0 + 1) < 8 ? (k0 + 1) : 8;
            float va = w1[ka * F1 + n1];
            float vb = w1[kb * F1 + n1];
            Bw1[t][c].x = (k0 < NUM_IN) ? va : 0.0f;
            Bw1[t][c].y = (k0 + 1 < NUM_IN) ? vb : 0.0f;
        }
    }

    float sum_t[2] = {0.f, 0.f};
    float sq_t[2]  = {0.f, 0.f};

    int nwaves = gridDim.x * (blockDim.x >> 5);
    int wid    = blockIdx.x * (blockDim.x >> 5) + (tid >> 5);

    for (int tile = wid; tile < NTILE; tile += nwaves) {
        long pid = (long)tile * 16 + ml;
        const float* pp = points + pid * NUM_IN;

        v2f A1[3];
        A1[0].x = pp[ko];       A1[0].y = pp[ko + 1];
        A1[1].x = pp[ko + 4];   A1[1].y = pp[ko + 5];
        A1[2].x = hi ? 0.0f : pp[8];
        A1[2].y = 0.0f;

        v8f h[2];
#pragma unroll
        for (int t = 0; t < 2; ++t) h[t] = splat8(bias1[t]);
#pragma unroll
        for (int c = 0; c < 3; ++c)
#pragma unroll
            for (int t = 0; t < 2; ++t)
                h[t] = __builtin_amdgcn_wmma_f32_16x16x4_f32(
                    false, A1[c], false, Bw1[t][c], (short)0, h[t], false, false);

#pragma unroll
        for (int t = 0; t < 2; ++t)
#pragma unroll
            for (int r = 0; r < 8; ++r) {
                float v = h[t][r];
                sum_t[t] += v;
                sq_t[t]  = fmaf(v, v, sq_t[t]);
            }
    }

#pragma unroll
    for (int t = 0; t < 2; ++t) {
        sum_t[t] += __shfl_xor(sum_t[t], 16, 32);
        sq_t[t]  += __shfl_xor(sq_t[t], 16, 32);
    }
    if (!hi) {
#pragma unroll
        for (int t = 0; t < 2; ++t) {
            atomicAdd(&sstat[t * 16 + ml], sum_t[t]);
            atomicAdd(&sstat[F1 + t * 16 + ml], sq_t[t]);
        }
    }
    __syncthreads();
    if (tid < 2 * F1) atomicAdd(&ws[WS_SUM1 + tid], sstat[tid]);
}

// ---------------------------------------------------------------------
// 2) finalize BN -> scale/shift (BN folded: y = h*sc + sh)
// ---------------------------------------------------------------------
__global__ void finalize1_kernel(float* __restrict__ ws,
                                 const float* __restrict__ g,
                                 const float* __restrict__ be) {
    int j = threadIdx.x;                  // 32 threads
    float inv_n = 1.0f / (float)NPTS;
    float mean = ws[WS_SUM1 + j] * inv_n;
    float var  = fmaf(-mean, mean, ws[WS_SQ1 + j] * inv_n);
    float sc   = g[j] * rsqrtf(var + BN_EPS);
    ws[WS_SC1 + j] = sc;
    ws[WS_SH1 + j] = fmaf(-mean, sc, be[j]);
}

__global__ void finalize2_kernel(float* __restrict__ ws,
                                 const float* __restrict__ g,
                                 const float* __restrict__ be) {
    int j = threadIdx.x;                  // 64 threads
    float inv_n = 1.0f / (float)NPTS;
    float mean = ws[WS_SUM2 + j] * inv_n;
    float var  = fmaf(-mean, mean, ws[WS_SQ2 + j] * inv_n);
    float sc   = g[j] * rsqrtf(var + BN_EPS);
    ws[WS_SC2 + j] = sc;
    ws[WS_SH2 + j] = fmaf(-mean, sc, be[j]);
}

// ---------------------------------------------------------------------
// 3/5) fully-WMMA fused MLP.
//  layer1: A = x tile (16x12), B = w1e (BN1 folded), C = b1e splat,
//          ReLU applied on D, staged into a per-wave LDS tile (16x32)
//          to transpose M-on-VGPR -> M-on-lane.
//  layer2: A read back from LDS (ds_load_b64 per K-chunk), B = w2
//          (BN2 folded for the scatter pass), C = bias splat.
//  STATS=true : accumulate layer-2 pre-activation sum/sumsq
//  STATS=false: ReLU + segment scatter-max via int atomicMax
// ---------------------------------------------------------------------
template <bool STATS>
__global__ void __launch_bounds__(256, 2)
mlp_kernel(const float* __restrict__ points,
           const float* __restrict__ w1,
           const float* __restrict__ b1,
           const float* __restrict__ w2,
           const float* __restrict__ b2,
           const int*   __restrict__ idx32,  // int64 indices viewed as int pairs
           float* __restrict__ out,
           float* __restrict__ ws) {
    __shared__ float stage[8][16 * F1];     // per-wave feat1 tile (2KB each)
    __shared__ float sstat[2 * F2];

    int tid  = threadIdx.x;
    int lane = tid & 31;
    int ml   = lane & 15;
    bool hi  = lane >= 16;
    int hi2  = hi ? 2 : 0;
    int ko   = hi2;
    int hi8  = hi ? 8 : 0;
    float* st = stage[tid >> 5];

    if (STATS && tid < 2 * F2) sstat[tid] = 0.0f;
    __syncthreads();

    // ---- loop-invariant per-lane preloads ----
    // layer-1 B fragments with BN1 folded: w1e = w1*sc1, b1e = b1*sc1+sh1
    v2f Bw1[2][3];
    float bias1[2];
#pragma unroll
    for (int t = 0; t < 2; ++t) {
        int n1 = t * 16 + ml;
        float s1 = ws[WS_SC1 + n1];
        bias1[t] = fmaf(b1[n1], s1, ws[WS_SH1 + n1]);
#pragma unroll
        for (int c = 0; c < 3; ++c) {
            int k0 = 4 * c + hi2;
            int ka = k0 < 8 ? k0 : 8;
            int kb = (k0 + 1) < 8 ? (k0 + 1) : 8;
            float va = w1[ka * F1 + n1] * s1;
            float vb = w1[kb * F1 + n1] * s1;
            Bw1[t][c].x = (k0 < NUM_IN) ? va : 0.0f;
            Bw1[t][c].y = (k0 + 1 < NUM_IN) ? vb : 0.0f;
        }
    }

    // layer-2 B fragments (BN2 folded in the scatter pass)
    v2f Breg[4][8];
    float bias2[4];
#pragma unroll
    for (int t = 0; t < 4; ++t) {
        int n = t * 16 + ml;
        float s2 = STATS ? 1.0f : ws[WS_SC2 + n];
        bias2[t] = STATS ? b2[n] : fmaf(b2[n], s2, ws[WS_SH2 + n]);
#pragma unroll
        for (int c = 0; c < 8; ++c) {
            int k0 = 4 * c + hi2;
            Breg[t][c].x = w2[k0 * F2 + n] * s2;
            Breg[t][c].y = w2[(k0 + 1) * F2 + n] * s2;
        }
    }

    float sum_t[4] = {0.f, 0.f, 0.f, 0.f};
    float sq_t[4]  = {0.f, 0.f, 0.f, 0.f};

    int nwaves = gridDim.x * (blockDim.x >> 5);
    int wid    = blockIdx.x * (blockDim.x >> 5) + (tid >> 5);

    for (int tile = wid; tile < NTILE; tile += nwaves) {
        long pid = (long)tile * 16 + ml;
        const float* pp = points + pid * NUM_IN;

        // ---- layer 1 (WMMA, BN1 folded, K padded 9->12) ----
        v2f A1[3];
        A1[0].x = pp[ko];       A1[0].y = pp[ko + 1];
        A1[1].x = pp[ko + 4];   A1[1].y = pp[ko + 5];
        A1[2].x = hi ? 0.0f : pp[8];
        A1[2].y = 0.0f;

        v8f h[2];
#pragma unroll
        for (int t = 0; t < 2; ++t) h[t] = splat8(bias1[t]);
#pragma unroll
        for (int c = 0; c < 3; ++c)
#pragma unroll
            for (int t = 0; t < 2; ++t)
                h[t] = __builtin_amdgcn_wmma_f32_16x16x4_f32(
                    false, A1[c], false, Bw1[t][c], (short)0, h[t], false, false);

        // ReLU + transpose through wave-private LDS tile (M-on-VGPR -> M-on-lane)
#pragma unroll
        for (int t = 0; t < 2; ++t)
#pragma unroll
            for (int r = 0; r < 8; ++r)
                st[(r + hi8) * F1 + t * 16 + ml] = fmaxf(h[t][r], 0.0f);

        // ---- layer 2 (WMMA, 8 K-chunks x 4 N-tiles) ----
        v8f acc[4];
#pragma unroll
        for (int t = 0; t < 4; ++t) acc[t] = splat8(bias2[t]);
#pragma unroll
        for (int c = 0; c < 8; ++c) {
            v2f A2 = *(const v2f*)(st + ml * F1 + 4 * c + ko);
#pragma unroll
            for (int t = 0; t < 4; ++t)
                acc[t] = __builtin_amdgcn_wmma_f32_16x16x4_f32(
                    false, A2, false, Breg[t][c], (short)0, acc[t], false, false);
        }

        if (STATS) {
#pragma unroll
            for (int t = 0; t < 4; ++t)
#pragma unroll
                for (int r = 0; r < 8; ++r) {
                    float v = acc[t][r];
                    sum_t[t] += v;
                    sq_t[t]  = fmaf(v, v, sq_t[t]);
                }
        } else {
            int segv = idx32[2 * pid];   // low dword of this point's int64 segment id
#pragma unroll
            for (int r = 0; r < 8; ++r) {
                int s = __shfl(segv, r + hi8, 32);
#pragma unroll
                for (int t = 0; t < 4; ++t) {
                    float v = fmaxf(acc[t][r], 0.0f);
                    if (v > 0.0f) {
                        int n = t * 16 + ml;
                        atomicMax((int*)out + ((long)s * F2 + n), __float_as_int(v));
                    }
                }
            }
        }
    }

    if (STATS) {
#pragma unroll
        for (int t = 0; t < 4; ++t) {
            sum_t[t] += __shfl_xor(sum_t[t], 16, 32);
            sq_t[t]  += __shfl_xor(sq_t[t], 16, 32);
        }
        if (!hi) {
#pragma unroll
            for (int t = 0; t < 4; ++t) {
                atomicAdd(&sstat[t * 16 + ml], sum_t[t]);
                atomicAdd(&sstat[F2 + t * 16 + ml], sq_t[t]);
            }
        }
        __syncthreads();
        if (tid < 2 * F2) atomicAdd(&ws[WS_SUM2 + tid], sstat[tid]);
    }
}

// ---------------------------------------------------------------------
extern "C" void kernel_launch(void* const* d_in, const int* in_sizes, int n_in,
                              void* d_out, int out_size, void* d_ws, size_t ws_size,
                              hipStream_t stream) {
    (void)in_sizes; (void)n_in; (void)ws_size;
    const float* points = (const float*)d_in[0];
    const int*   idx32  = (const int*)d_in[1];   // int64 data, read low dwords
    const float* w1     = (const float*)d_in[3];
    const float* b1     = (const float*)d_in[4];
    const float* g1     = (const float*)d_in[5];
    const float* be1    = (const float*)d_in[6];
    const float* w2     = (const float*)d_in[7];
    const float* b2     = (const float*)d_in[8];
    const float* g2     = (const float*)d_in[9];
    const float* be2    = (const float*)d_in[10];
    float* out = (float*)d_out;
    float* ws  = (float*)d_ws;

    int zb = (out_size + 255) / 256;
    zero_kernel<<<zb, 256, 0, stream>>>(out, out_size, ws);

    stats1_kernel<<<2048, 256, 0, stream>>>(points, w1, b1, ws);
    finalize1_kernel<<<1, 32, 0, stream>>>(ws, g1, be1);

    mlp_kernel<true><<<2048, 256, 0, stream>>>(points, w1, b1, w2, b2, nullptr, out, ws);
    finalize2_kernel<<<1, 64, 0, stream>>>(ws, g2, be2);

    mlp_kernel<false><<<2048, 256, 0, stream>>>(points, w1, b1, w2, b2, idx32, out, ws);
}